// QModule_9139690406427
// MI455X (gfx1250) — compile-verified
//
#include <hip/hip_runtime.h>

// B=64, N=1024, C=192, G=4.  out = (ori_activ, activ) concatenated flat.
// Pure streaming elementwise kernel: 50.3 MB in, 100.6 MB out -> HBM-bound
// (~6.5 us floor at 23.3 TB/s). No matmul structure -> no WMMA by design.

typedef float v4f __attribute__((ext_vector_type(4)));

#define GROUPS   4
#define CHANNELS 192
#define BDIM     192   // 6 wave32 waves; 192 = 4*48 -> channel quad is loop-invariant

__global__ __launch_bounds__(BDIM) void qmodule_fused_kernel(
    const float* __restrict__ in,       // (B,N,C) f32
    const float* __restrict__ gr,       // (G,)    f32
    const float* __restrict__ alpha,    // (G,C)   f32
    float* __restrict__ out_ori,        // (B,N,C) f32
    float* __restrict__ out_q,          // (B,N,C) f32
    long long n_vec)                    // B*N*C/4
{
    __shared__ float s_sw[GROUPS][CHANNELS];

    const int tid = threadIdx.x;

    // ---- per-channel softmax over the 4 groups (once per block, trivial cost)
    if (tid < CHANNELS) {
        float a0 = alpha[0 * CHANNELS + tid];
        float a1 = alpha[1 * CHANNELS + tid];
        float a2 = alpha[2 * CHANNELS + tid];
        float a3 = alpha[3 * CHANNELS + tid];
        float m  = fmaxf(fmaxf(a0, a1), fmaxf(a2, a3));
        float e0 = expf(a0 - m);
        float e1 = expf(a1 - m);
        float e2 = expf(a2 - m);
        float e3 = expf(a3 - m);
        float inv = 1.0f / (e0 + e1 + e2 + e3);
        s_sw[0][tid] = e0 * inv;
        s_sw[1][tid] = e1 * inv;
        s_sw[2][tid] = e2 * inv;
        s_sw[3][tid] = e3 * inv;
    }

    // ---- group scalars (uniform; 4 dword loads per thread, cached)
    float r[GROUPS], invb[GROUPS], bq[GROUPS];
#pragma unroll
    for (int g = 0; g < GROUPS; ++g) {
        float rv = gr[g];
        r[g]    = rv;
        bq[g]   = rv * (1.0f / 255.0f);                 // b = r/255
        invb[g] = (rv != 0.0f) ? (255.0f / rv) : 0.0f;  // 1/b (0 kills q when r==0)
    }

    __syncthreads();

    // ---- hoist the 16 weights + 16 pre-scaled weights into VGPRs.
    // stride = gridDim*192 is a multiple of 48, so c4 never changes per thread.
    const int c4 = (tid % 48) * 4;
    float sw[GROUPS][4], bsw[GROUPS][4];
#pragma unroll
    for (int g = 0; g < GROUPS; ++g)
#pragma unroll
        for (int j = 0; j < 4; ++j) {
            sw[g][j]  = s_sw[g][c4 + j];
            bsw[g][j] = bq[g] * sw[g][j];   // fold b into the q-path weight
        }

    const v4f* __restrict__ vin  = (const v4f*)in;
    v4f* __restrict__ vori       = (v4f*)out_ori;
    v4f* __restrict__ vq         = (v4f*)out_q;

    long long idx = (long long)blockIdx.x * BDIM + tid;
    const long long stride = (long long)gridDim.x * BDIM;

    for (; idx < n_vec; idx += stride) {
        v4f x = __builtin_nontemporal_load(vin + idx);   // global_load_b128, TH=NT

        v4f accO = {0.0f, 0.0f, 0.0f, 0.0f};
        v4f accQ = {0.0f, 0.0f, 0.0f, 0.0f};

#pragma unroll
        for (int g = 0; g < GROUPS; ++g) {
#pragma unroll
            for (int j = 0; j < 4; ++j) {
                // clamp(x,0,r) == 0.5*(|x|-|x-r|+r) for r>=0  -> one v_med3_f32
                float ori = __builtin_amdgcn_fmed3f(x[j], 0.0f, r[g]);
                // round-half-even matches jnp.round -> v_rndne_f32
                float qn  = rintf(ori * invb[g]);
                accO[j] = fmaf(ori, sw[g][j],  accO[j]);
                accQ[j] = fmaf(qn,  bsw[g][j], accQ[j]);
            }
        }

        __builtin_nontemporal_store(accO, vori + idx);   // global_store_b128, TH=NT
        __builtin_nontemporal_store(accQ, vq   + idx);
    }
}

extern "C" void kernel_launch(void* const* d_in, const int* in_sizes, int n_in,
                              void* d_out, int out_size, void* d_ws, size_t ws_size,
                              hipStream_t stream) {
    (void)n_in; (void)d_ws; (void)ws_size;

    const float* in    = (const float*)d_in[0];  // inputs (B,N,C)
    const float* gr    = (const float*)d_in[1];  // groups_range1 (G,)
    const float* alpha = (const float*)d_in[2];  // alpha_activ (G,C)

    float* out_ori = (float*)d_out;                    // ori_activ first
    float* out_q   = (float*)d_out + (out_size / 2);   // activ second

    const long long n_elem = (long long)in_sizes[0];   // B*N*C (multiple of 4)
    const long long n_vec  = n_elem / 4;

    // 2048 blocks * 192 threads -> exactly 8 float4 per thread for the
    // reference shape; grid-stride keeps it correct for any size.
    long long need = (n_vec + BDIM - 1) / BDIM;
    int blocks = (int)((need < 2048) ? need : 2048);
    if (blocks < 1) blocks = 1;

    qmodule_fused_kernel<<<blocks, BDIM, 0, stream>>>(in, gr, alpha,
                                                      out_ori, out_q, n_vec);
}